// BalancedSupervisedContrastiveRegressionLoss_71880572666388
// MI455X (gfx1250) — compile-verified
//
#include <hip/hip_runtime.h>
#include <math.h>

// ---------------------------------------------------------------------------
// BalancedSupervisedContrastiveRegressionLoss for MI455X (gfx1250)
// N=2048, D=512, VOCAB=121, T=0.07, LABEL_OFFSET=40
//
// Math restructuring (exact, not approximate):
//   A[i,j]      = (exp(S[i,j]/T - max_i) + 1e-5) * tw_i * tw_j,  S = P P^T
//   d[i,j]      = |c_i - c_j|   (c = class index, 0..120)
//   denom[i,j]  = sum_k A[j,k] * (d[i,k] >= d[i,j])
//               = F[c_i, j],  F[c,j] = sum_{c'} C[j,c'] * (|c-c'| >= |c-c_j|)
//   where C[j,c] = sum_{k: cls(k)=c} A[j,k]
//   loss = (1/N) * 1/(N-1+1e-5) * sum_i [ -(rowlogsumA_i - logA_ii)
//                                         + G[c_i] - logF[c_i, i] ]
//   with G[c] = sum_j log F[c,j]
// ---------------------------------------------------------------------------

typedef __attribute__((ext_vector_type(2))) float v2f;
typedef __attribute__((ext_vector_type(8))) float v8f;

#define NN 2048
#define DD 512
#define NVOC 121
#define CPAD 128
#define LOFF 40
#define INV_T (1.0f / 0.07f)

#define KC 128            // K-chunk staged in LDS
#define BSTRIDE (KC + 4)  // 132 floats: 16B-aligned rows, bank-conflict-free

// workspace layout (bytes)
#define OFF_S      ((size_t)0)                       // NN*NN*4     = 16 MB
#define OFF_C      (OFF_S      + (size_t)NN*NN*4)    // NN*CPAD*4   = 1 MB
#define OFF_LOGF   (OFF_C      + (size_t)NN*CPAD*4)  // CPAD*NN*4   = 1 MB
#define OFF_RMAX   (OFF_LOGF   + (size_t)CPAD*NN*4)  // NN*4
#define OFF_TW     (OFF_RMAX   + (size_t)NN*4)
#define OFF_CLS    (OFF_TW     + (size_t)NN*4)
#define OFF_RLS    (OFF_CLS    + (size_t)NN*4)
#define OFF_DIAG   (OFF_RLS    + (size_t)NN*4)
#define OFF_FDIAG  (OFF_DIAG   + (size_t)NN*4)
#define OFF_G      (OFF_FDIAG  + (size_t)NN*4)       // CPAD*4

// ---------------------------------------------------------------- kernel 0
__global__ void k_setup(const int* __restrict__ targets,
                        const float* __restrict__ weights,
                        float* __restrict__ tw, int* __restrict__ cls) {
  int i = blockIdx.x * blockDim.x + threadIdx.x;
  if (i < NN) {
    int c = targets[i] - LOFF;
    cls[i] = c;
    tw[i] = weights[c];
  }
}

// ---------------------------------------------------------------- kernel 1
// S = P @ P^T via V_WMMA_F32_16X16X4_F32, B tile staged in LDS.
// Block = 256 threads = 8 waves, computes 128x64 of S. Grid = (NN/128, NN/64).
// All 8 waves share the 64 B-rows -> stage them once per block in LDS
// (64 x KC floats, row stride 132 -> conflict-free ds_load_b64 fragments).
//
// f32 16x16x4 fragment layout (ISA 7.12.2):
//   A (16x4): lane&15 = M row; K = (lane>>4)*2 + v   (v = VGPR 0/1)
//   B (4x16): lane&15 = N col; same K mapping
//   C/D     : lane&15 = N; VGPR v -> M = v + 8*(lane>>4)
__global__ __launch_bounds__(256) void k_gemm(const float* __restrict__ P,
                                              float* __restrict__ S) {
  __shared__ float Bs[64][BSTRIDE];  // ~33 KB

  const int tid  = threadIdx.x;
  const int wave = tid >> 5;
  const int lane = tid & 31;
  const int half = lane >> 4;
  const int r    = lane & 15;

  const int mrow  = blockIdx.x * 128 + wave * 16 + r;
  const int nbase = blockIdx.y * 64;

  const float* __restrict__ ap = P + (size_t)mrow * DD;

  v8f acc0 = {}, acc1 = {}, acc2 = {}, acc3 = {};

  for (int kc = 0; kc < DD; kc += KC) {
    // stage B[nbase .. nbase+63][kc .. kc+KC) : 2048 float4, 8 per thread.
    // Loads issued into registers before the barrier -> overlap with the
    // previous chunk's WMMA compute.
    float4 stg[8];
#pragma unroll
    for (int t = 0; t < 8; ++t) {
      const int idx = tid + t * 256;
      const int row = idx >> 5;        // 0..63
      const int c4  = (idx & 31) * 4;  // 0..124
      stg[t] = *(const float4*)(P + (size_t)(nbase + row) * DD + kc + c4);
    }
    __syncthreads();  // previous chunk fully consumed
#pragma unroll
    for (int t = 0; t < 8; ++t) {
      const int idx = tid + t * 256;
      const int row = idx >> 5;
      const int c4  = (idx & 31) * 4;
      *(float4*)&Bs[row][c4] = stg[t];  // ds_store_b128, 16B-aligned
    }
    __syncthreads();  // staging visible to all waves

#pragma unroll 4
    for (int kk = 0; kk < KC; kk += 4) {
      const int kg = kc + kk + half * 2;  // global K for A fragment
      const int kl = kk + half * 2;       // LDS K for B fragments
      v2f a  = { ap[kg], ap[kg + 1] };
      v2f b0 = { Bs[ 0 + r][kl], Bs[ 0 + r][kl + 1] };
      v2f b1 = { Bs[16 + r][kl], Bs[16 + r][kl + 1] };
      v2f b2 = { Bs[32 + r][kl], Bs[32 + r][kl + 1] };
      v2f b3 = { Bs[48 + r][kl], Bs[48 + r][kl + 1] };
      acc0 = __builtin_amdgcn_wmma_f32_16x16x4_f32(false, a, false, b0, (short)0, acc0, false, false);
      acc1 = __builtin_amdgcn_wmma_f32_16x16x4_f32(false, a, false, b1, (short)0, acc1, false, false);
      acc2 = __builtin_amdgcn_wmma_f32_16x16x4_f32(false, a, false, b2, (short)0, acc2, false, false);
      acc3 = __builtin_amdgcn_wmma_f32_16x16x4_f32(false, a, false, b3, (short)0, acc3, false, false);
    }
  }

  const int mb = blockIdx.x * 128 + wave * 16 + half * 8;
#pragma unroll
  for (int v = 0; v < 8; ++v) {
    float* row = S + (size_t)(mb + v) * NN + nbase + r;
    row[0]  = acc0[v];
    row[16] = acc1[v];
    row[32] = acc2[v];
    row[48] = acc3[v];
  }
}

// ---------------------------------------------------------------- kernel 2
// rowmax[i] = max_j S[i,j] / T   (deterministic LDS tree)
__global__ __launch_bounds__(256) void k_rowmax(const float* __restrict__ S,
                                                float* __restrict__ rmax) {
  const int i = blockIdx.x, tid = threadIdx.x;
  __shared__ float red[256];
  float m = -3.4e38f;
  for (int j = tid; j < NN; j += 256) m = fmaxf(m, S[(size_t)i * NN + j]);
  red[tid] = m;
  __syncthreads();
  for (int s = 128; s > 0; s >>= 1) {
    if (tid < s) red[tid] = fmaxf(red[tid], red[tid + s]);
    __syncthreads();
  }
  if (tid == 0) rmax[i] = red[0] * INV_T;
}

// ---------------------------------------------------------------- kernel 3
// One block per row i. Computes A[i,j] on the fly from S, accumulates:
//   C[i,c]      (class-bucketed row sums; deterministic fixed-order scan)
//   rowlogsum_i = sum_j log A[i,j]
//   diag_i      = log A[i,i]
__global__ __launch_bounds__(256) void k_arow(const float* __restrict__ S,
                                              const float* __restrict__ rmax,
                                              const float* __restrict__ tw,
                                              const int* __restrict__ cls,
                                              float* __restrict__ C,
                                              float* __restrict__ rls,
                                              float* __restrict__ diag) {
  const int i = blockIdx.x, tid = threadIdx.x;
  __shared__ float avals[256];
  __shared__ int   jcl[256];
  __shared__ float cacc[CPAD];
  __shared__ float red[256];

  if (tid < CPAD) cacc[tid] = 0.0f;
  __syncthreads();

  const float twi = tw[i];
  const float mi  = rmax[i];
  float sumlog = 0.0f;

  for (int base = 0; base < NN; base += 256) {
    const int j = base + tid;
    const float s = S[(size_t)i * NN + j];
    float a = expf(s * INV_T - mi) + 1e-5f;
    a *= twi * tw[j];
    const float lg = logf(a);
    sumlog += lg;
    if (j == i) diag[i] = lg;
    avals[tid] = a;
    jcl[tid]   = cls[j];
    __syncthreads();
    if (tid < NVOC) {  // fixed-order scan -> deterministic
      float acc = 0.0f;
      for (int t = 0; t < 256; ++t) acc += (jcl[t] == tid) ? avals[t] : 0.0f;
      cacc[tid] += acc;
    }
    __syncthreads();
  }

  if (tid < CPAD) C[(size_t)i * CPAD + tid] = cacc[tid];
  red[tid] = sumlog;
  __syncthreads();
  for (int s = 128; s > 0; s >>= 1) {
    if (tid < s) red[tid] += red[tid + s];
    __syncthreads();
  }
  if (tid == 0) rls[i] = red[0];
}

// ---------------------------------------------------------------- kernel 4
// One block (128 threads) per sample j: F[c,j] for all anchor classes c.
//   logF[c*NN + j] = log F[c,j];  Fdiag[j] = log F[cls(j), j]
__global__ __launch_bounds__(128) void k_fden(const float* __restrict__ C,
                                              const int* __restrict__ cls,
                                              float* __restrict__ logF,
                                              float* __restrict__ Fdiag) {
  const int j = blockIdx.x;
  const int c = threadIdx.x;  // 0..127
  __shared__ float crow[CPAD];
  crow[c] = C[(size_t)j * CPAD + c];  // entries >= NVOC are zero (k_arow wrote them)
  __syncthreads();
  if (c < NVOC) {
    const int cj = cls[j];
    const int v  = abs(c - cj);
    float acc = 0.0f;
    for (int cp = 0; cp < NVOC; ++cp)
      acc += (abs(c - cp) >= v) ? crow[cp] : 0.0f;
    const float lf = logf(acc);
    logF[(size_t)c * NN + j] = lf;
    if (c == cj) Fdiag[j] = lf;
  }
}

// ---------------------------------------------------------------- kernel 5
// G[c] = sum_j logF[c, j]
__global__ __launch_bounds__(256) void k_gsum(const float* __restrict__ logF,
                                              float* __restrict__ G) {
  const int c = blockIdx.x, tid = threadIdx.x;
  __shared__ float red[256];
  float s = 0.0f;
  for (int j = tid; j < NN; j += 256) s += logF[(size_t)c * NN + j];
  red[tid] = s;
  __syncthreads();
  for (int sh = 128; sh > 0; sh >>= 1) {
    if (tid < sh) red[tid] += red[tid + sh];
    __syncthreads();
  }
  if (tid == 0) G[c] = red[0];
}

// ---------------------------------------------------------------- kernel 6
__global__ __launch_bounds__(256) void k_final(const float* __restrict__ rls,
                                               const float* __restrict__ diag,
                                               const float* __restrict__ G,
                                               const float* __restrict__ Fdiag,
                                               const int* __restrict__ cls,
                                               float* __restrict__ out) {
  const int tid = threadIdx.x;
  __shared__ float red[256];
  float s = 0.0f;
  for (int i = tid; i < NN; i += 256)
    s += -(rls[i] - diag[i]) + G[cls[i]] - Fdiag[i];
  red[tid] = s;
  __syncthreads();
  for (int sh = 128; sh > 0; sh >>= 1) {
    if (tid < sh) red[tid] += red[tid + sh];
    __syncthreads();
  }
  if (tid == 0)
    out[0] = red[0] / ((float)(NN - 1) + 1e-5f) / (float)NN;
}

// ---------------------------------------------------------------------------
extern "C" void kernel_launch(void* const* d_in, const int* in_sizes, int n_in,
                              void* d_out, int out_size, void* d_ws, size_t ws_size,
                              hipStream_t stream) {
  const float* P       = (const float*)d_in[0];  // [N, D]
  const int*   targets = (const int*)d_in[1];    // [N]
  const float* weights = (const float*)d_in[2];  // [VOCAB]
  float* out = (float*)d_out;

  char* ws = (char*)d_ws;
  float* S     = (float*)(ws + OFF_S);
  float* C     = (float*)(ws + OFF_C);
  float* logF  = (float*)(ws + OFF_LOGF);
  float* rmax  = (float*)(ws + OFF_RMAX);
  float* tw    = (float*)(ws + OFF_TW);
  int*   cls   = (int*)  (ws + OFF_CLS);
  float* rls   = (float*)(ws + OFF_RLS);
  float* diag  = (float*)(ws + OFF_DIAG);
  float* Fdiag = (float*)(ws + OFF_FDIAG);
  float* G     = (float*)(ws + OFF_G);

  k_setup<<<dim3((NN + 255) / 256), dim3(256), 0, stream>>>(targets, weights, tw, cls);
  k_gemm<<<dim3(NN / 128, NN / 64), dim3(256), 0, stream>>>(P, S);
  k_rowmax<<<dim3(NN), dim3(256), 0, stream>>>(S, rmax);
  k_arow<<<dim3(NN), dim3(256), 0, stream>>>(S, rmax, tw, cls, C, rls, diag);
  k_fden<<<dim3(NN), dim3(128), 0, stream>>>(C, cls, logF, Fdiag);
  k_gsum<<<dim3(NVOC), dim3(256), 0, stream>>>(logF, G);
  k_final<<<dim3(1), dim3(256), 0, stream>>>(rls, diag, G, Fdiag, cls, out);
}